// PrimeHilbertEncoder_60979945668679
// MI455X (gfx1250) — compile-verified
//
#include <hip/hip_runtime.h>
#include <math.h>

typedef __attribute__((ext_vector_type(2))) float v2f;
typedef __attribute__((ext_vector_type(8))) float v8f;
typedef int v4i __attribute__((__vector_size__(4 * sizeof(int))));

#define NTOK     65536
#define KDIM     768
#define NDIM     67
#define NPAD     80          // 5 N-tiles of 16
#define MAXP     512

#define MT       2           // M-tiles (of 16 tokens) per block
#define MROWS    (MT * 16)   // 32 tokens per block
#define AS       772         // padded A row stride (floats): conflict-free ds_load_b64
#define KB       32          // K chunk per LDS stage
#define WS       36          // padded W chunk row stride (floats)
#define NCHUNK   (KDIM / KB) // 24 (even)
#define NTHREADS 160         // 5 wave32 waves
#define WCH_F4   ((NPAD * KB / 4) / NTHREADS)  // 4 float4 per thread per chunk

#define WLDS_F   (NPAD * WS)         // 2880 floats per W buffer
#define ADYN_BYTES (MROWS * AS * 4)  // 98816 B dynamic LDS for A tile

// ---- gfx1250 async-to-LDS path (guarded; fallback = plain copies) ----
#if defined(__has_builtin)
#if __has_builtin(__builtin_amdgcn_global_load_async_to_lds_b128) && \
    __has_builtin(__builtin_amdgcn_s_wait_asynccnt)
#define USE_ASYNC_LDS 1
#endif
#endif
#ifndef USE_ASYNC_LDS
#define USE_ASYNC_LDS 0
#endif

#if USE_ASYNC_LDS
#define GLOBAL_AS __attribute__((address_space(1)))
#define LDS_AS    __attribute__((address_space(3)))
__device__ __forceinline__ void async_copy16(const float* g, float* l) {
  __builtin_amdgcn_global_load_async_to_lds_b128(
      (GLOBAL_AS v4i*)(g), (LDS_AS v4i*)(l), 0, 0);
}
__device__ __forceinline__ void async_drain() {
  __builtin_amdgcn_s_wait_asynccnt(0);
}
#else
__device__ __forceinline__ void async_copy16(const float* g, float* l) {
  *(float4*)l = *(const float4*)g;
}
__device__ __forceinline__ void async_drain() {}
#endif

// ---------------------------------------------------------------------------
// Kernel 1: constant prime-sinusoid position table [512, 67] into workspace.
// ---------------------------------------------------------------------------
__global__ void build_pos_table(float* __restrict__ pt) {
  int idx = blockIdx.x * blockDim.x + threadIdx.x;
  if (idx >= MAXP * NDIM) return;
  int pos = idx / NDIM;
  int e   = idx - pos * NDIM;
  const int primes[5] = {7, 11, 13, 17, 19};
  int off = 0;
  float v = 0.0f;
  #pragma unroll
  for (int i = 0; i < 5; ++i) {
    int p = primes[i];
    if (e < off + p) {
      int j = e - off;
      int sin_dim = (p >> 1) + (p & 1);
      float phase = (float)(pos % p) / (float)p;
      float fpos  = (float)pos;
      float neg   = -logf(10000.0f) / (float)p;
      if (j < sin_dim) v = sinf(fpos * expf((float)j * neg) + phase);
      else             v = cosf(fpos * expf((float)(j - sin_dim) * neg) + phase);
      break;
    }
    off += p;
  }
  pt[idx] = v;
}

// ---------------------------------------------------------------------------
// Kernel 2: gather + fused GEMM (V_WMMA_F32_16X16X4_F32), 32 tokens / block,
//   async-to-LDS staging, distinct ping/pong W buffers, 2 acc tiles / wave.
// ---------------------------------------------------------------------------
__launch_bounds__(NTHREADS)
__global__ void prime_proj_kernel(const int*   __restrict__ ids,
                                  const int*   __restrict__ poss,
                                  const float* __restrict__ emb,
                                  const float* __restrict__ W,
                                  const float* __restrict__ bias,
                                  const float* __restrict__ postab,
                                  float*       __restrict__ out) {
  __shared__ float Wb0[WLDS_F];        // distinct objects -> AA can prove
  __shared__ float Wb1[WLDS_F];        // async writes don't alias WMMA reads
  __shared__ int   sIds[MROWS];
  __shared__ int   sPos[MROWS];
  extern __shared__ float Alds[];      // MROWS x AS (dynamic, 96.5 KB)

  const int t    = threadIdx.x;
  const int wave = t >> 5;             // 0..4 -> N tile
  const int lane = t & 31;
  const int half = lane >> 4;
  const int mr   = lane & 15;
  const int tok0 = blockIdx.x * MROWS;

  if (t < MROWS) {
    sIds[t] = ids[tok0 + t];
    int p = poss[tok0 + t];
    sPos[t] = p < 0 ? 0 : (p > MAXP - 1 ? MAXP - 1 : p);
  }
  __syncthreads();

  // Per-thread W-staging coordinates (n clamped: rows >= 67 feed masked cols).
  int wdst[WCH_F4];
  const float* wsrc[WCH_F4];
  #pragma unroll
  for (int u = 0; u < WCH_F4; ++u) {
    int idx = t + u * NTHREADS;
    int n   = idx >> 3;                 // KB/4 = 8 float4 per row
    int c4  = (idx & 7) * 4;
    int nc  = n < NDIM ? n : NDIM - 1;
    wdst[u] = n * WS + c4;
    wsrc[u] = W + (size_t)nc * KDIM + c4;
  }

  // Gather 32 x 768 fp32 embedding rows into LDS (async b128 per lane).
  for (int idx = t; idx < MROWS * (KDIM / 4); idx += NTHREADS) {
    int row = idx / (KDIM / 4);
    int c4  = (idx - row * (KDIM / 4)) * 4;
    async_copy16(emb + (size_t)sIds[row] * KDIM + c4, Alds + row * AS + c4);
  }
  // Stage W chunk 0 into ping buffer.
  #pragma unroll
  for (int u = 0; u < WCH_F4; ++u) async_copy16(wsrc[u], Wb0 + wdst[u]);
  async_drain();
  __syncthreads();

  v8f acc0 = {}, acc1 = {};
  const float* A0  = Alds + mr * AS + 2 * half;          // tile-0 rows
  const float* A1  = Alds + (16 + mr) * AS + 2 * half;   // tile-1 rows
  const int    bof = (wave * 16 + mr) * WS + 2 * half;
  const float* B0  = Wb0 + bof;
  const float* B1  = Wb1 + bof;

#define COMPUTE_CHUNK(Bp, ka)                                                  \
  _Pragma("unroll") for (int j = 0; j < KB / 4; ++j) {                         \
    v2f b  = *(const v2f*)((Bp) + 4 * j);                                      \
    v2f a0 = *(const v2f*)(A0 + (ka) + 4 * j);                                 \
    v2f a1 = *(const v2f*)(A1 + (ka) + 4 * j);                                 \
    acc0 = __builtin_amdgcn_wmma_f32_16x16x4_f32(false, a0, false, b,          \
                                                 (short)0, acc0, false, false);\
    acc1 = __builtin_amdgcn_wmma_f32_16x16x4_f32(false, a1, false, b,          \
                                                 (short)0, acc1, false, false);\
  }

#define STAGE_W(buf, kc)                                                       \
  _Pragma("unroll") for (int u = 0; u < WCH_F4; ++u)                           \
      async_copy16(wsrc[u] + (kc), (buf) + wdst[u]);

  for (int cc = 0; cc < NCHUNK; cc += 2) {
    // Even phase: prefetch chunk cc+1 into pong while computing on ping.
    STAGE_W(Wb1, (cc + 1) * KB);
    COMPUTE_CHUNK(B0, cc * KB);
    async_drain();
    __syncthreads();
    // Odd phase: prefetch chunk cc+2 into ping while computing on pong.
    if (cc + 2 < NCHUNK) { STAGE_W(Wb0, (cc + 2) * KB); }
    COMPUTE_CHUNK(B1, (cc + 1) * KB);
    async_drain();
    __syncthreads();
  }

  // Epilogue: acc[r] holds (m = r + 8*half, n = wave*16 + mr).
  const int n = wave * 16 + mr;
  if (n < NDIM) {
    const float bv = bias[n];
    #pragma unroll
    for (int rr = 0; rr < 8; ++rr) {
      int m = rr + 8 * half;
      out[(size_t)(tok0 + m) * NDIM + n] =
          acc0[rr] + bv + postab[(size_t)sPos[m] * NDIM + n];
      out[(size_t)(tok0 + 16 + m) * NDIM + n] =
          acc1[rr] + bv + postab[(size_t)sPos[16 + m] * NDIM + n];
    }
  }
}

// ---------------------------------------------------------------------------
extern "C" void kernel_launch(void* const* d_in, const int* in_sizes, int n_in,
                              void* d_out, int out_size, void* d_ws, size_t ws_size,
                              hipStream_t stream) {
  (void)in_sizes; (void)n_in; (void)out_size; (void)ws_size;
  const int*   ids    = (const int*)d_in[0];
  const int*   poss   = (const int*)d_in[1];
  const float* emb    = (const float*)d_in[2];
  const float* W      = (const float*)d_in[3];
  const float* bias   = (const float*)d_in[4];
  float*       out    = (float*)d_out;
  float*       postab = (float*)d_ws;   // 512*67*4 = 137 KB scratch

  build_pos_table<<<(MAXP * NDIM + 255) / 256, 256, 0, stream>>>(postab);
  prime_proj_kernel<<<NTOK / MROWS, NTHREADS, ADYN_BYTES, stream>>>(
      ids, poss, emb, W, bias, postab, out);
}